// BiologicallyInformedLoss_19533511262777
// MI455X (gfx1250) — compile-verified
//
#include <hip/hip_runtime.h>
#include <hip/hip_bf16.h>

// ---------------------------------------------------------------------------
// BiologicallyInformedLoss for MI455X (gfx1250, wave32).
//
//  K1 pos_kernel      : branchless streaming softmax over (B,L,NC) logits;
//                       per-block LDS-atomic histograms -> global atomics.
//  K2 groupsum_wmma   : RSCU codon->amino-group sum as one-hot GEMM
//                       (64 x 68 x 32) on V_WMMA_F32_16X16X4_F32; operands
//                       staged branchlessly in LDS, tight ds_load+wmma loop.
//  K3 finalize        : per-sequence CAI / RSCU-KL / gc / mfe / pause terms.
// ---------------------------------------------------------------------------

#define B_    64
#define L_    8192
#define NC_   65
#define KPAD  68            // K padded to 17 * 4 for 16x16x4 WMMA steps
#define NGPAD 32            // amino groups padded 22 -> 32 (two N tiles)
#define EPSF  1e-8f

// per-b workspace record (floats):
//  [0..64]    cnt_pred     [65..129]  obs_pred
//  [130..194] cnt_tgt      [195..259] obs_tgt
//  [260] logw_pred  [261] logw_tgt  [262] mask_cnt  [263] gc_sum  [264] pause_sum
#define PB_STRIDE 268
#define PB_BASE   16        // ws[0]=ce_sum, ws[1]=valid_cnt, rest pad
#define GS_BASE   (PB_BASE + B_ * PB_STRIDE)          // 17168
#define WS_FLOATS (GS_BASE + 2 * B_ * NGPAD)

// Codon -> amino-group id (from AA64; '*'=1,'A'=2,...,'Y'=21; 0=pad), padded
// to KPAD with -1 sentinels so the one-hot B matrix is zero in the tail.
__constant__ int d_GROUP68[KPAD] = {
  0,
  6, 6,11,11,17,17,17,17,21,21, 1, 1, 3, 3, 1,20,
 11,11,11,11,14,14,14,14, 8, 8,15,15,16,16,16,16,
  9, 9, 9,12,18,18,18,18,13,13,10,10,17,17,16,16,
 19,19,19,19, 2, 2, 2, 2, 4, 4, 5, 5, 7, 7, 7, 7,
 -1,-1,-1
};

// Synonymous-codon counts per codon id
__constant__ float d_NSYN[NC_] = {
  0.f,
  2.f,2.f,6.f,6.f,6.f,6.f,6.f,6.f,2.f,2.f,3.f,3.f,2.f,2.f,3.f,1.f,
  6.f,6.f,6.f,6.f,4.f,4.f,4.f,4.f,2.f,2.f,2.f,2.f,6.f,6.f,6.f,6.f,
  3.f,3.f,3.f,1.f,4.f,4.f,4.f,4.f,2.f,2.f,2.f,2.f,6.f,6.f,6.f,6.f,
  4.f,4.f,4.f,4.f,4.f,4.f,4.f,4.f,2.f,2.f,2.f,2.f,4.f,4.f,4.f,4.f
};

__device__ __forceinline__ bool is_coding(int c) {
  return (c != 0) & (c != 11) & (c != 12) & (c != 15);
}

// ---------------------------------------------------------------------------
// K1: streaming per-position kernel (branchless softmax core)
// ---------------------------------------------------------------------------
__global__ void __launch_bounds__(256)
pos_kernel(const float* __restrict__ logits,
           const float* __restrict__ wmat,        // (5, 65)
           const int*   __restrict__ tgt,
           const int*   __restrict__ aa,
           const int*   __restrict__ species,
           const unsigned char* __restrict__ mask,
           const float* __restrict__ gc,
           const float* __restrict__ pause,
           float*       __restrict__ ws)
{
  __shared__ float sh[272];
  const int tid = threadIdx.x;
  for (int i = tid; i < 272; i += 256) sh[i] = 0.f;
  __syncthreads();

  const int b   = blockIdx.y;
  const int pos = blockIdx.x * 256 + tid;
  const int idx = b * L_ + pos;
  const float* lrow = logits + (size_t)idx * NC_;

  const int t = tgt[idx];

  // Load row into registers (fully unrolled -> load clause, no scratch),
  // capturing the target logit branchlessly against the constant index.
  float xv[NC_];
  float tl = 0.f;
  #pragma unroll
  for (int c = 0; c < NC_; ++c) {
    const float x = lrow[c];
    xv[c] = x;
    tl = (c == t) ? x : tl;
  }

  // Branchless max + first-max argmax.
  float m = xv[0];
  int   pid = 0;
  #pragma unroll
  for (int c = 1; c < NC_; ++c) {
    const bool gt = xv[c] > m;
    pid = gt ? c : pid;
    m   = gt ? xv[c] : m;
  }

  // Single exp-sum pass.
  float s = 0.f;
  #pragma unroll
  for (int c = 0; c < NC_; ++c) s += __expf(xv[c] - m);

  const float nll = __logf(s) + m - tl;

  const bool mk = mask[idx] != 0;
  const int  a  = aa[idx];
  const int  sp = species[b];

  // cross-entropy over (target != 0); mask not applied (matches reference)
  const float v = (t != 0) ? 1.f : 0.f;
  atomicAdd(&sh[260], nll * v);
  atomicAdd(&sh[261], v);

  if (mk) {
    atomicAdd(&sh[262], __logf(fmaxf(wmat[sp * NC_ + pid], EPSF)));
    atomicAdd(&sh[263], __logf(fmaxf(wmat[sp * NC_ + t  ], EPSF)));
    atomicAdd(&sh[264], 1.f);
    if (pid > 0) {
      atomicAdd(&sh[pid], 1.f);
      if (a > 2) atomicAdd(&sh[65 + pid], 1.f);
    }
    if (t > 0) {
      atomicAdd(&sh[130 + t], 1.f);
      if (a > 2) atomicAdd(&sh[195 + t], 1.f);
    }
  }
  atomicAdd(&sh[265], gc[idx]);
  atomicAdd(&sh[266], pause[idx]);
  __syncthreads();

  float* pb = ws + PB_BASE + b * PB_STRIDE;
  for (int i = tid; i < 260; i += 256)
    if (sh[i] != 0.f) atomicAdd(&pb[i], sh[i]);
  if (tid == 0) { atomicAdd(&ws[0], sh[260]); atomicAdd(&ws[1], sh[261]); }
  if (tid == 1) atomicAdd(&pb[260], sh[262]);
  if (tid == 2) atomicAdd(&pb[261], sh[263]);
  if (tid == 3) atomicAdd(&pb[262], sh[264]);
  if (tid == 4) atomicAdd(&pb[263], sh[265]);
  if (tid == 5) atomicAdd(&pb[264], sh[266]);
}

// ---------------------------------------------------------------------------
// K2: RSCU group sums as one-hot GEMM on V_WMMA_F32_16X16X4_F32.
//     D[b][g] = sum_c obs_counts[b][c] * (GROUP_IDS[c] == g)
//     Operands staged branchlessly in LDS; inner loop = ds_load + wmma only.
//     One wave, EXEC all ones at every WMMA.
// ---------------------------------------------------------------------------
typedef float v2f __attribute__((ext_vector_type(2)));
typedef float v8f __attribute__((ext_vector_type(8)));

__global__ void __launch_bounds__(32)
groupsum_wmma_kernel(float* __restrict__ ws)
{
  __shared__ float ldsA[2][B_][KPAD];     // obs_counts (pred/tgt), zero-padded K
  __shared__ float ldsB[2][KPAD][16];     // one-hot codon->group, per N tile
  const int lane = threadIdx.x;
  const int half = lane >> 4;
  const int lm   = lane & 15;

  // Stage A: obs_counts[b][c] = coding && obs>0 ? cnt : 0  (branchless selects)
  for (int which = 0; which < 2; ++which) {
    const int cntoff = which ? 130 : 0;
    const int obsoff = which ? 195 : 65;
    for (int bb = 0; bb < B_; ++bb) {
      const float* pb = ws + PB_BASE + bb * PB_STRIDE;
      for (int k = lane; k < KPAD; k += 32) {
        const int  kc   = (k < NC_) ? k : 0;
        const float cnt = pb[cntoff + kc];
        const float ob  = pb[obsoff + kc];
        const bool keep = (k < NC_) & is_coding(kc) & (ob > 0.f);
        ldsA[which][bb][k] = keep ? cnt : 0.f;
      }
    }
  }
  // Stage B: one-hot membership matrix per N tile (group ids 0..31 padded)
  for (int i = lane; i < 2 * KPAD * 16; i += 32) {
    const int nt  = i / (KPAD * 16);
    const int rem = i - nt * (KPAD * 16);
    const int k   = rem >> 4;
    const int col = rem & 15;
    ldsB[nt][k][col] = (d_GROUP68[k] == nt * 16 + col) ? 1.f : 0.f;
  }
  __syncthreads();

  for (int which = 0; which < 2; ++which) {
    float* gs = ws + GS_BASE + which * (B_ * NGPAD);
    for (int mt = 0; mt < 4; ++mt) {
      const float* arow = &ldsA[which][mt * 16 + lm][0];
      for (int nt = 0; nt < 2; ++nt) {
        v8f acc = {};
        #pragma unroll
        for (int kt = 0; kt < 17; ++kt) {
          // ISA A layout: K split across lane halves, 2 K-values per VGPR pair
          const int ka = kt * 4 + half * 2;
          v2f a, bm;
          a.x  = arow[ka];
          a.y  = arow[ka + 1];
          bm.x = ldsB[nt][ka][lm];
          bm.y = ldsB[nt][ka + 1][lm];
          acc = __builtin_amdgcn_wmma_f32_16x16x4_f32(
              false, a, false, bm, (short)0, acc, false, false);
        }
        // C/D layout: VGPR r -> M = r + 8*half, N = lane&15
        #pragma unroll
        for (int r = 0; r < 8; ++r)
          gs[(mt * 16 + r + half * 8) * NGPAD + (nt * 16 + lm)] = acc[r];
      }
    }
  }
}

// ---------------------------------------------------------------------------
// K3: finalize — per-sequence terms, reduce, scalar out
// ---------------------------------------------------------------------------
__device__ __forceinline__ float rscu_val(const float* pb, const float* gs,
                                          int c, int cntoff, int obsoff) {
  if (c == 0) return 0.f;
  const bool obs = is_coding(c) && (pb[obsoff + c] > 0.f);
  const float tot = gs[d_GROUP68[c]];
  return (obs && tot > 0.f) ? pb[cntoff + c] * d_NSYN[c] / fmaxf(tot, 1.f) : 0.f;
}

__global__ void __launch_bounds__(64)
finalize_kernel(const float* __restrict__ ref,      // (5, 65)
                const float* __restrict__ mfe,      // (B,)
                const int*   __restrict__ species,  // (B,)
                const float* __restrict__ ws,
                float*       __restrict__ out)
{
  __shared__ float red[B_];
  const int b = threadIdx.x;

  const float* pb  = ws + PB_BASE + b * PB_STRIDE;
  const float* gsP = ws + GS_BASE + b * NGPAD;
  const float* gsT = ws + GS_BASE + B_ * NGPAD + b * NGPAD;
  const float* rr  = ref + species[b] * NC_;

  const float mcnt = fmaxf(pb[262], 1.f);
  const float caiP = __expf(pb[260] / mcnt);
  const float caiT = __expf(pb[261] / mcnt);
  const float cai_term = fmaxf(caiT - caiP, 0.f);

  float Sp = 0.f, St = 0.f;
  for (int c = 0; c < NC_; ++c) {
    Sp += rscu_val(pb, gsP, c, 0, 65) + EPSF;
    St += 0.7f * rscu_val(pb, gsT, c, 130, 195) + 0.3f * rr[c] + EPSF;
  }
  float kl = 0.f;
  for (int c = 0; c < NC_; ++c) {
    const float p = (rscu_val(pb, gsP, c, 0, 65) + EPSF) / Sp;
    const float t = (0.7f * rscu_val(pb, gsT, c, 130, 195) + 0.3f * rr[c] + EPSF) / St;
    kl += t * (__logf(t) - __logf(p));
  }

  const float gcm = pb[263] * (1.f / (float)L_) - 0.5f;
  const float pm  = pb[264] * (1.f / (float)L_) - 0.1f;
  const float mf  = mfe[b] + 20.f;

  red[b] = 0.4f * cai_term + 0.3f * kl + 0.1f * gcm * gcm
         + 0.15f * mf * mf + 0.1f * pm * pm;
  __syncthreads();

  if (b == 0) {
    float acc = 0.f;
    for (int i = 0; i < B_; ++i) acc += red[i];
    const float ce = ws[0] / fmaxf(ws[1], 1.f);
    out[0] = ce + acc * (1.f / (float)B_);
  }
}

// ---------------------------------------------------------------------------
extern "C" void kernel_launch(void* const* d_in, const int* in_sizes, int n_in,
                              void* d_out, int out_size, void* d_ws, size_t ws_size,
                              hipStream_t stream) {
  const float* logits = (const float*)d_in[0];
  const float* wmat   = (const float*)d_in[1];
  const float* refd   = (const float*)d_in[2];
  const float* gc     = (const float*)d_in[3];
  const float* mfe    = (const float*)d_in[4];
  const float* pause  = (const float*)d_in[5];
  const int*   tgt    = (const int*)d_in[6];
  const int*   aa     = (const int*)d_in[7];
  const int*   spec   = (const int*)d_in[8];
  const unsigned char* mask = (const unsigned char*)d_in[9];
  float* out = (float*)d_out;
  float* ws  = (float*)d_ws;

  (void)in_sizes; (void)n_in; (void)out_size; (void)ws_size;

  hipMemsetAsync(ws, 0, (size_t)WS_FLOATS * sizeof(float), stream);

  dim3 grid(L_ / 256, B_);
  pos_kernel<<<grid, 256, 0, stream>>>(logits, wmat, tgt, aa, spec, mask, gc, pause, ws);
  groupsum_wmma_kernel<<<1, 32, 0, stream>>>(ws);
  finalize_kernel<<<1, 64, 0, stream>>>(refd, mfe, spec, ws, out);
}